// Renderer_13752485282321
// MI455X (gfx1250) — compile-verified
//
#include <hip/hip_runtime.h>
#include <math.h>

#define WAVES_PER_BLOCK 8
#define S 128
#define FARD 1e10f

// ---- CDNA5 async global->LDS path (device-compile only) ----
#if defined(__HIP_DEVICE_COMPILE__) && defined(__gfx1250__) && \
    __has_builtin(__builtin_amdgcn_global_load_async_to_lds_b128)
#define HAVE_ASYNC_LDS 1
#else
#define HAVE_ASYNC_LDS 0
#endif

#if HAVE_ASYNC_LDS
typedef int v4i_t __attribute__((ext_vector_type(4)));
typedef __attribute__((address_space(1))) v4i_t as1_v4i;
typedef __attribute__((address_space(3))) v4i_t as3_v4i;

__device__ __forceinline__ void async_copy_b128(const float* g, float* l) {
  // 16B per lane, DMA into LDS, tracked on ASYNCcnt
  __builtin_amdgcn_global_load_async_to_lds_b128((as1_v4i*)g, (as3_v4i*)l, 0, 0);
}
__device__ __forceinline__ void wait_async0() {
#if __has_builtin(__builtin_amdgcn_s_wait_asynccnt)
  __builtin_amdgcn_s_wait_asynccnt(0);
#else
  asm volatile("s_wait_asynccnt 0" ::: "memory");
#endif
}
#endif

__device__ __forceinline__ float fsigmoid(float x) {
  return 1.0f / (1.0f + __expf(-x));
}

__global__ __launch_bounds__(256) void nerf_render_sample_kernel(
    const float* __restrict__ raw,     // [N,128,4]
    const float* __restrict__ z_vals,  // [N,128]
    const float* __restrict__ rays_d,  // [N,3]
    float* __restrict__ out,           // 262*N floats, concatenated
    int n_rays) {
  __shared__ float s_raw[WAVES_PER_BLOCK][S * 4];  // 2KB/wave
  __shared__ float s_z[WAVES_PER_BLOCK][S];
  __shared__ float s_w[WAVES_PER_BLOCK][S];
  __shared__ float s_cdf[WAVES_PER_BLOCK][S];      // 127 used

  const int lane = threadIdx.x & 31;
  const int wv   = threadIdx.x >> 5;
  const int ray  = blockIdx.x * WAVES_PER_BLOCK + wv;
  const bool active = (ray < n_rays);
  const size_t n = (size_t)n_rays;

  float* rawsh = s_raw[wv];
  float* zsh   = s_z[wv];
  float* wsh   = s_w[wv];
  float* csh   = s_cdf[wv];

  // ---------- Stage 1: DMA ray data into LDS ----------
  if (active) {
    const float* graw = raw + (size_t)ray * (S * 4);
    const float* gz   = z_vals + (size_t)ray * S;
#if HAVE_ASYNC_LDS
#pragma unroll
    for (int t = 0; t < 4; ++t) {
      const int c = lane + 32 * t;  // 16B chunk index, 128 chunks = 2KB
      async_copy_b128(graw + c * 4, rawsh + c * 4);
    }
    async_copy_b128(gz + lane * 4, zsh + lane * 4);
    wait_async0();
#else
#pragma unroll
    for (int t = 0; t < 4; ++t) {
      const int c = lane + 32 * t;
      *(float4*)(rawsh + c * 4) = *(const float4*)(graw + c * 4);
    }
    *(float4*)(zsh + lane * 4) = *(const float4*)(gz + lane * 4);
#endif
  }
  __syncthreads();

  // ---------- Stage 2: ray march (lane owns samples 4*lane .. 4*lane+3) ----------
  if (active) {
    const float dx = rays_d[(size_t)ray * 3 + 0];
    const float dy = rays_d[(size_t)ray * 3 + 1];
    const float dz = rays_d[(size_t)ray * 3 + 2];
    const float nrm = sqrtf(dx * dx + dy * dy + dz * dz);

    float a[4], q[4], zv[4], rr[4], gg[4], bb[4];
#pragma unroll
    for (int j = 0; j < 4; ++j) {
      const int s = 4 * lane + j;
      const float4 rw = *(const float4*)(rawsh + s * 4);
      zv[j] = zsh[s];
      const float dist = ((s < S - 1) ? (zsh[s + 1] - zv[j]) : FARD) * nrm;
      const float sig = fmaxf(rw.w, 0.0f);
      a[j] = 1.0f - __expf(-sig * dist);
      q[j] = 1.0f - a[j] + 1e-10f;
      rr[j] = fsigmoid(rw.x);
      gg[j] = fsigmoid(rw.y);
      bb[j] = fsigmoid(rw.z);
    }

    // exclusive cumulative product across 128 samples:
    // per-lane product + Kogge-Stone scan across 32 lanes
    float inc = q[0] * q[1] * q[2] * q[3];
#pragma unroll
    for (int off = 1; off < 32; off <<= 1) {
      const float t = __shfl_up(inc, off, 32);
      if (lane >= off) inc *= t;
    }
    float excl = __shfl_up(inc, 1, 32);
    if (lane == 0) excl = 1.0f;

    const float t0 = excl;
    const float t1 = t0 * q[0];
    const float t2 = t1 * q[1];
    const float t3 = t2 * q[2];
    const float w0 = a[0] * t0;
    const float w1 = a[1] * t1;
    const float w2 = a[2] * t2;
    const float w3 = a[3] * t3;

    // keep weights in LDS for the pdf stage + coalesced store to out
    wsh[4 * lane + 0] = w0;
    wsh[4 * lane + 1] = w1;
    wsh[4 * lane + 2] = w2;
    wsh[4 * lane + 3] = w3;
    float* outw = out + 5 * n + (size_t)ray * S;
    *(float4*)(outw + 4 * lane) = make_float4(w0, w1, w2, w3);

    // weighted reductions
    float acc = w0 + w1 + w2 + w3;
    float dep = w0 * zv[0] + w1 * zv[1] + w2 * zv[2] + w3 * zv[3];
    float cr = w0 * rr[0] + w1 * rr[1] + w2 * rr[2] + w3 * rr[3];
    float cg = w0 * gg[0] + w1 * gg[1] + w2 * gg[2] + w3 * gg[3];
    float cb = w0 * bb[0] + w1 * bb[1] + w2 * bb[2] + w3 * bb[3];
#pragma unroll
    for (int off = 16; off > 0; off >>= 1) {
      acc += __shfl_xor(acc, off, 32);
      dep += __shfl_xor(dep, off, 32);
      cr  += __shfl_xor(cr, off, 32);
      cg  += __shfl_xor(cg, off, 32);
      cb  += __shfl_xor(cb, off, 32);
    }
    if (lane == 0) {
      out[(size_t)ray * 3 + 0] = cr;
      out[(size_t)ray * 3 + 1] = cg;
      out[(size_t)ray * 3 + 2] = cb;
      out[3 * n + ray] = 1.0f / fmaxf(1e-10f, dep / acc);  // disp
      out[4 * n + ray] = acc;                              // acc
      out[133 * n + ray] = dep;                            // depth
    }
  }
  __syncthreads();

  // ---------- Stage 3a: build normalized CDF (127 entries) in LDS ----------
  if (active) {
    // pdf over weights[1:-1] -> 126 values; lane handles i = 4l..4l+3
    float pv[4];
    float ps = 0.0f;
#pragma unroll
    for (int j = 0; j < 4; ++j) {
      const int i = 4 * lane + j;
      pv[j] = (i < S - 2) ? (wsh[i + 1] + 1e-5f) : 0.0f;
      ps += pv[j];
    }
    float cinc = ps;
#pragma unroll
    for (int off = 1; off < 32; off <<= 1) {
      const float t = __shfl_up(cinc, off, 32);
      if (lane >= off) cinc += t;
    }
    const float total = __shfl(cinc, 31, 32);
    float cex = __shfl_up(cinc, 1, 32);
    if (lane == 0) cex = 0.0f;
    const float invT = 1.0f / total;  // total >= 126e-5 > 0

    float run = cex;
#pragma unroll
    for (int j = 0; j < 4; ++j) {
      const int i = 4 * lane + j;
      run += pv[j];
      if (i < S - 2) csh[i + 1] = run * invT;
    }
    if (lane == 0) csh[0] = 0.0f;
  }
  __syncthreads();

  // ---------- Stage 3b: inverse-CDF sampling, 4 u's per lane ----------
  if (active) {
    float zs[4];
#pragma unroll
    for (int j = 0; j < 4; ++j) {
      const int k = 4 * lane + j;
      const float u = (float)k * (1.0f / 127.0f);  // linspace(0,1,128)
      // searchsorted(csh[0..126], u, side='right')
      int lo = 0, hi = 127;
      while (lo < hi) {
        const int mid = (lo + hi) >> 1;
        if (csh[mid] <= u) lo = mid + 1; else hi = mid;
      }
      int below = lo - 1; if (below < 0) below = 0;
      int above = lo;     if (above > 126) above = 126;
      const float c0 = csh[below];
      const float c1 = csh[above];
      const float b0 = 0.5f * (zsh[below] + zsh[below + 1]);
      const float b1 = 0.5f * (zsh[above] + zsh[above + 1]);
      float denom = c1 - c0;
      denom = (denom < 1e-5f) ? 1.0f : denom;
      const float t = (u - c0) / denom;
      zs[j] = b0 + t * (b1 - b0);
    }
    float* outz = out + 134 * n + (size_t)ray * S;
    *(float4*)(outz + 4 * lane) = make_float4(zs[0], zs[1], zs[2], zs[3]);
  }
}

extern "C" void kernel_launch(void* const* d_in, const int* in_sizes, int n_in,
                              void* d_out, int out_size, void* d_ws, size_t ws_size,
                              hipStream_t stream) {
  const float* raw    = (const float*)d_in[0];
  const float* z_vals = (const float*)d_in[1];
  const float* rays_d = (const float*)d_in[2];
  float* out = (float*)d_out;

  const int n_rays = in_sizes[2] / 3;  // rays_d is [N,3]
  const int blocks = (n_rays + WAVES_PER_BLOCK - 1) / WAVES_PER_BLOCK;
  nerf_render_sample_kernel<<<blocks, 256, 0, stream>>>(raw, z_vals, rays_d, out, n_rays);
}